// minGRU2_43035572306725
// MI455X (gfx1250) — compile-verified
//
#include <hip/hip_runtime.h>
#include <cstdint>

namespace {
constexpr int L     = 8192;          // sequence length
constexpr int NTH   = 256;           // threads per block (8 waves, wave32)
constexpr int CHUNK = L / NTH;       // 32 timesteps per thread
constexpr int PAD   = L + 2 * (L / 32);  // skew-2 layout: 8704 floats (34 KB)
constexpr int HCH   = 512;           // input channels
constexpr int OCH   = 256;           // output channels

__device__ __forceinline__ int skew(int pos) {
  // +2 floats every 32: preserves 8B blocks (async b64 alignment) and gives
  // stride-34 chunk walks -> 32 distinct LDS banks across 32 lanes.
  return pos + ((pos >> 5) << 1);
}

__device__ __forceinline__ float laddexp(float a, float b) {
  float m = fmaxf(a, b);
  if (m == -__builtin_inff()) return -__builtin_inff();
  return m + __logf(1.0f + __expf(fminf(a, b) - m));
}
}  // namespace

__global__ __launch_bounds__(NTH) void mingru_bidir_scan_kernel(
    const float* __restrict__ x, float* __restrict__ out) {
  __shared__ float As[PAD];   // h row -> la -> cumA -> exp(log_h)
  __shared__ float Bs[PAD];   // gate row -> lv -> local scan value
  __shared__ float sA[NTH];   // chunk aggregate: sum of la
  __shared__ float sV[NTH];   // chunk aggregate: log scan value

  const int t   = threadIdx.x;
  const int seq = blockIdx.x;          // b*256 + dir*128 + c
  const int b   = seq >> 8;
  const int r   = seq & 255;
  const bool rev = (r >= 128);
  const int c   = r & 127;

  const size_t hRow = (size_t)b * HCH * L + (size_t)(rev ? 256 + c : c) * L;
  const size_t gRow = hRow + (size_t)128 * L;
  const size_t oRow = (size_t)b * OCH * L + (size_t)(rev ? 128 + c : c) * L;

  // ---- Stage 1: CDNA5 async global->LDS copy of both rows (ASYNCcnt) ----
  {
    const unsigned long long hp = (unsigned long long)(uintptr_t)(x + hRow);
    const unsigned long long gp = (unsigned long long)(uintptr_t)(x + gRow);
    // low 32 bits of the flat address of a __shared__ object = LDS offset
    const uint32_t aB = (uint32_t)(uintptr_t)(void*)As;
    const uint32_t bB = (uint32_t)(uintptr_t)(void*)Bs;
#pragma unroll
    for (int it = 0; it < (L / 2) / NTH; ++it) {   // 16 iters of 8B chunks
      const int ck = it * NTH + t;                 // 8B chunk id (pos = 2*ck)
      const uint32_t goff = (uint32_t)ck << 3;                 // global byte off
      const uint32_t doff = goff + (uint32_t)((ck >> 4) << 3); // skewed LDS off
      const uint32_t da = aB + doff;
      const uint32_t db = bB + doff;
      asm volatile("global_load_async_to_lds_b64 %0, %1, %2"
                   :: "v"(da), "v"(goff), "s"(hp) : "memory");
      asm volatile("global_load_async_to_lds_b64 %0, %1, %2"
                   :: "v"(db), "v"(goff), "s"(gp) : "memory");
    }
    asm volatile("s_wait_asynccnt 0" ::: "memory");
  }
  __syncthreads();

  // ---- Stage 2: elementwise log-space transform (in place, coalesced) ----
  for (int pos = t; pos < L; pos += NTH) {
    const int p = skew(pos);
    const float h = As[p];
    const float g = Bs[p];
    // softplus(g) once: la = -sp, logsigmoid(g) = g - sp
    const float sp = fmaxf(g, 0.0f) + __logf(1.0f + __expf(-fabsf(g)));
    const float ht = (h < 0.0f) ? h : __logf(1.0f + h);
    As[p] = -sp;            // log_coeffs
    Bs[p] = (g - sp) + ht;  // log_values
  }
  __syncthreads();

  // ---- Stage 3: per-thread sequential scan of a contiguous time chunk ----
  float cum = 0.0f;
  float V   = -__builtin_inff();
  {
    const int base = t * CHUNK;
#pragma unroll 4
    for (int i = 0; i < CHUNK; ++i) {
      const int idx = base + i;                       // time index
      const int pos = rev ? (L - 1 - idx) : idx;      // memory index
      const int p = skew(pos);
      const float la = As[p];
      const float lv = Bs[p];
      cum += la;                                      // running sum(log_coeffs)
      V = laddexp(la + V, lv);                        // running logcumsumexp
      As[p] = cum;
      Bs[p] = V;
    }
  }
  sA[t] = cum;
  sV[t] = V;
  __syncthreads();

  // ---- Stage 4: Hillis-Steele scan of chunk aggregates (log-space compose) --
#pragma unroll
  for (int off = 1; off < NTH; off <<= 1) {
    float pa = 0.0f, pv = -__builtin_inff();
    const bool has = (t >= off);
    if (has) { pa = sA[t - off]; pv = sV[t - off]; }
    __syncthreads();
    if (has) {
      const float ca = sA[t];
      const float cv = sV[t];
      sA[t] = pa + ca;                    // earlier-then-later composition
      sV[t] = laddexp(ca + pv, cv);
    }
    __syncthreads();
  }
  const float prefV = (t == 0) ? -__builtin_inff() : sV[t - 1];

  // ---- Stage 5: apply incoming state, exponentiate ----
  {
    const int base = t * CHUNK;
#pragma unroll 4
    for (int i = 0; i < CHUNK; ++i) {
      const int idx = base + i;
      const int pos = rev ? (L - 1 - idx) : idx;
      const int p = skew(pos);
      const float lh = laddexp(As[p] + prefV, Bs[p]);
      As[p] = __expf(lh);
    }
  }
  __syncthreads();

  // ---- Stage 6: coalesced float2 stores (flip handled by index mapping) ----
  for (int v = t; v < L / 2; v += NTH) {
    const int pos = v * 2;
    const int p = skew(pos);          // p even; pairs contiguous under skew
    float2 o;
    o.x = As[p];
    o.y = As[p + 1];
    *reinterpret_cast<float2*>(out + oRow + pos) = o;
  }
}

extern "C" void kernel_launch(void* const* d_in, const int* in_sizes, int n_in,
                              void* d_out, int out_size, void* d_ws, size_t ws_size,
                              hipStream_t stream) {
  const float* x = (const float*)d_in[0];
  float* out = (float*)d_out;
  const int B = in_sizes[0] / (HCH * L);      // 4
  const int nseq = B * OCH;                   // B * 2 dirs * 128 channels = 1024
  mingru_bidir_scan_kernel<<<dim3(nseq), dim3(NTH), 0, stream>>>(x, out);
}